// MultiHeadAttention_46325517254760
// MI455X (gfx1250) — compile-verified
//
#include <hip/hip_runtime.h>
#include <math.h>
#include <stdint.h>

typedef float v2f __attribute__((ext_vector_type(2)));
typedef float v8f __attribute__((ext_vector_type(8)));

#define NNODE 512
#define CCH   16
#define TT    15
#define TP    16           // T+1
#define HH    32
#define DDIM  32
#define NTP   (NNODE*TP)   // 8192

// ---- workspace layout (floats) ----
enum : size_t {
  OFF_A1 = 0,
  OFF_B1 = OFF_A1 + (size_t)NNODE*TP*HH,
  OFF_A2 = OFF_B1 + (size_t)NNODE*TP*HH,
  OFF_B2 = OFF_A2 + (size_t)NNODE*TP*DDIM,
  OFF_Q  = OFF_B2 + (size_t)NNODE*TP*DDIM,
  OFF_M2 = OFF_Q  + (size_t)NNODE*HH,
  OFF_M3 = OFF_M2 + (size_t)NNODE*TP,
  OFF_M1 = OFF_M3 + (size_t)NNODE*TP,
  OFF_M4 = OFF_M1 + (size_t)NNODE*NTP,
  OFF_L1 = OFF_M4 + (size_t)NNODE*NTP,
  OFF_L2 = OFF_L1 + (size_t)NNODE*NNODE,
  OFF_R1 = OFF_L2 + (size_t)NNODE*NNODE,
  OFF_R2 = OFF_R1 + (size_t)NNODE*TP,
  OFF_T1 = OFF_R2 + (size_t)NNODE*TP,
  OFF_T2 = OFF_T1 + (size_t)NNODE*DDIM,
  OFF_END = OFF_T2 + (size_t)NNODE*DDIM
};

__device__ __forceinline__ v8f wmma_f32x4(v2f a, v2f b, v8f c) {
  // D = A(16x4,f32) x B(4x16,f32) + C(16x16,f32), wave32
  return __builtin_amdgcn_wmma_f32_16x16x4_f32(false, a, false, b, (short)0, c, false, false);
}

// CDNA5 async global->LDS copy (ASYNCcnt-tracked), per CDNA5 ISA.
// lds_off is the LDS byte offset (addr[31:0] of the generic LDS pointer).
__device__ __forceinline__ void async_load_f32_to_lds(uint32_t lds_off, const float* gsrc) {
  asm volatile("global_load_async_to_lds_b32 %0, %1, off"
               :: "v"(lds_off), "v"(gsrc) : "memory");
}
__device__ __forceinline__ void wait_asynccnt0() {
  asm volatile("s_wait_asynccnt 0x0" ::: "memory");
}

// ---------------------------------------------------------------------------
// 1) Per-node precompute: A1,B1,A2,B2 [N][16][32], Q [N][32], M2/M3 [N][16]
// ---------------------------------------------------------------------------
__global__ void precompute_k(const float* __restrict__ x,  const float* __restrict__ W1,
                             const float* __restrict__ W2, const float* __restrict__ W3,
                             float* __restrict__ A1, float* __restrict__ B1,
                             float* __restrict__ A2, float* __restrict__ B2,
                             float* __restrict__ Q,  float* __restrict__ M2,
                             float* __restrict__ M3)
{
  int n = blockIdx.x, tid = threadIdx.x;
  __shared__ float xs[CCH];
  __shared__ float A1s[TP*HH], B1s[TP*HH], Qs[HH];
  if (tid < CCH) xs[tid] = x[n*CCH + tid];
  __syncthreads();
  for (int idx = tid; idx < TP*HH; idx += 256) {   // idx = t*32 + h
    float a1 = 0.f, b1 = 0.f, a2 = 0.f, b2 = 0.f;
#pragma unroll
    for (int i = 0; i < CCH; ++i) {
      float xi = xs[i];
      a1 += xi * W1[i*TP*HH + idx];
      b1 += xi * W1[(i+CCH)*TP*HH + idx];
      a2 += xi * W2[i*TP*DDIM + idx];
      b2 += xi * W2[(i+CCH)*TP*DDIM + idx];
    }
    A1s[idx] = a1; B1s[idx] = b1;
    A1[(size_t)n*TP*HH + idx]   = a1;
    B1[(size_t)n*TP*HH + idx]   = b1;
    A2[(size_t)n*TP*DDIM + idx] = a2;
    B2[(size_t)n*TP*DDIM + idx] = b2;
  }
  if (tid < HH) {
    float q = 0.f;
#pragma unroll
    for (int i = 0; i < CCH; ++i) q += xs[i] * W3[i*HH + tid];
    Qs[tid] = q;
    Q[n*HH + tid] = q;
  }
  __syncthreads();
  if (tid < TP) {
    float m2 = 0.f, m3 = 0.f;
#pragma unroll
    for (int h = 0; h < HH; ++h) {
      m2 += Qs[h] * B1s[tid*HH + h];
      m3 += Qs[h] * A1s[tid*HH + h];
    }
    M2[n*TP + tid] = m2;
    M3[n*TP + tid] = m3;
  }
}

// ---------------------------------------------------------------------------
// 2) WMMA GEMM: M1 = Q(512x32) @ A1flat^T(32x8192); M4 = Q @ B1flat^T
//    one wave per 16x16 output tile; K=32 -> 8 x (16x16x4 f32 WMMA)
// ---------------------------------------------------------------------------
__global__ void gemm_qat_k(const float* __restrict__ Q,  const float* __restrict__ A1,
                           const float* __restrict__ B1, float* __restrict__ M1,
                           float* __restrict__ M4)
{
  const float* Bsrc = blockIdx.z ? B1 : A1;   // rows = (b,t) pairs, cols = h
  float* Dst        = blockIdx.z ? M4 : M1;
  int col0 = blockIdx.x * 16, a0 = blockIdx.y * 16;
  int l = threadIdx.x, m = l & 15, kh = l >> 4;
  v8f c = {};
#pragma unroll
  for (int k0 = 0; k0 < HH; k0 += 4) {
    int k = k0 + 2*kh;
    v2f a, b;
    a.x = Q[(a0 + m)*HH + k];      a.y = Q[(a0 + m)*HH + k + 1];
    b.x = Bsrc[(size_t)(col0 + m)*HH + k];
    b.y = Bsrc[(size_t)(col0 + m)*HH + k + 1];
    c = wmma_f32x4(a, b, c);
  }
  int ccol = col0 + m, rbase = a0 + 8*kh;
#pragma unroll
  for (int r = 0; r < 8; ++r)
    Dst[(size_t)(rbase + r)*NTP + ccol] = c[r];
}

// ---------------------------------------------------------------------------
// 3) logits1/logits2 from adj, M1..M4 (+ diagonal terms)
// ---------------------------------------------------------------------------
__global__ void logits_k(const float* __restrict__ adj, const float* __restrict__ M1,
                         const float* __restrict__ M2,  const float* __restrict__ M3,
                         const float* __restrict__ M4,  float* __restrict__ L1,
                         float* __restrict__ L2)
{
  int a = blockIdx.y;
  int b = blockIdx.x * 256 + threadIdx.x;
  const float* adjab = adj + ((size_t)a*NNODE + b)*TT;
  const float* adjba = adj + ((size_t)b*NNODE + a)*TT;
  const float* m1 = M1 + (size_t)a*NTP + b*TP;
  const float* m2 = M2 + a*TP;
  const float* m3 = M3 + b*TP;
  const float* m4 = M4 + (size_t)b*NTP + a*TP;
  float l1 = 0.f, l2 = 0.f;
#pragma unroll
  for (int t = 0; t < TT; ++t) {
    float w = adjab[t], v = adjba[t];
    l1 += w * m1[t] + v * m2[t];
    l2 += w * m3[t] + v * m4[t];
  }
  if (a == b) { l1 += m1[TT] + m2[TT]; l2 += m3[TT] + m4[TT]; }
  L1[(size_t)a*NNODE + b] = l1;
  L2[(size_t)a*NNODE + b] = l2;
}

// ---------------------------------------------------------------------------
// 4) softmax (row for s1, column for s2), in place
// ---------------------------------------------------------------------------
__global__ void softmax_row_k(float* __restrict__ L)
{
  int a = blockIdx.x, tid = threadIdx.x;
  __shared__ float red[256];
  float mx = -1e30f;
  for (int b = tid; b < NNODE; b += 256) mx = fmaxf(mx, L[(size_t)a*NNODE + b]);
  red[tid] = mx; __syncthreads();
  for (int s = 128; s > 0; s >>= 1) { if (tid < s) red[tid] = fmaxf(red[tid], red[tid+s]); __syncthreads(); }
  mx = red[0]; __syncthreads();
  float sum = 0.f;
  for (int b = tid; b < NNODE; b += 256) sum += __expf(L[(size_t)a*NNODE + b] - mx);
  red[tid] = sum; __syncthreads();
  for (int s = 128; s > 0; s >>= 1) { if (tid < s) red[tid] += red[tid+s]; __syncthreads(); }
  float inv = 1.0f / red[0];
  for (int b = tid; b < NNODE; b += 256)
    L[(size_t)a*NNODE + b] = __expf(L[(size_t)a*NNODE + b] - mx) * inv;
}

__global__ void softmax_col_k(float* __restrict__ L)
{
  int b = blockIdx.x, tid = threadIdx.x;
  __shared__ float red[256];
  float mx = -1e30f;
  for (int a = tid; a < NNODE; a += 256) mx = fmaxf(mx, L[(size_t)a*NNODE + b]);
  red[tid] = mx; __syncthreads();
  for (int s = 128; s > 0; s >>= 1) { if (tid < s) red[tid] = fmaxf(red[tid], red[tid+s]); __syncthreads(); }
  mx = red[0]; __syncthreads();
  float sum = 0.f;
  for (int a = tid; a < NNODE; a += 256) sum += __expf(L[(size_t)a*NNODE + b] - mx);
  red[tid] = sum; __syncthreads();
  for (int s = 128; s > 0; s >>= 1) { if (tid < s) red[tid] += red[tid+s]; __syncthreads(); }
  float inv = 1.0f / red[0];
  for (int a = tid; a < NNODE; a += 256)
    L[(size_t)a*NNODE + b] = __expf(L[(size_t)a*NNODE + b] - mx) * inv;
}

// ---------------------------------------------------------------------------
// 5) R1[n,t] = sum_b s1[n,b]*adj[b,n,t]  (R1[n,15]=s1[n,n])
//    R2[n,t] = sum_a s2[a,n]*adj[n,a,t]  (R2[n,15]=s2[n,n])
// ---------------------------------------------------------------------------
__global__ void compute_r_k(const float* __restrict__ adj, const float* __restrict__ s1,
                            const float* __restrict__ s2, float* __restrict__ R1,
                            float* __restrict__ R2)
{
  int n = blockIdx.x, tid = threadIdx.x;
  float r1[TT], r2[TT];
#pragma unroll
  for (int t = 0; t < TT; ++t) { r1[t] = 0.f; r2[t] = 0.f; }
  for (int o = tid; o < NNODE; o += 256) {
    float w1 = s1[(size_t)n*NNODE + o];
    float w2 = s2[(size_t)o*NNODE + n];
    const float* adj_on = adj + ((size_t)o*NNODE + n)*TT;
    const float* adj_no = adj + ((size_t)n*NNODE + o)*TT;
#pragma unroll
    for (int t = 0; t < TT; ++t) { r1[t] += w1 * adj_on[t]; r2[t] += w2 * adj_no[t]; }
  }
  __shared__ float red[256];
  for (int t = 0; t < TT; ++t) {
    red[tid] = r1[t]; __syncthreads();
    for (int s = 128; s > 0; s >>= 1) { if (tid < s) red[tid] += red[tid+s]; __syncthreads(); }
    if (tid == 0) R1[n*TP + t] = red[0];
    __syncthreads();
    red[tid] = r2[t]; __syncthreads();
    for (int s = 128; s > 0; s >>= 1) { if (tid < s) red[tid] += red[tid+s]; __syncthreads(); }
    if (tid == 0) R2[n*TP + t] = red[0];
    __syncthreads();
  }
  if (tid == 0) {
    R1[n*TP + TT] = s1[(size_t)n*NNODE + n];
    R2[n*TP + TT] = s2[(size_t)n*NNODE + n];
  }
}

// ---------------------------------------------------------------------------
// 6) Fused attend GEMM: OUT[row,d] = sum_{other,t} (s*adj)[row,(other,t)] * GB[other,t,d]
//                                    + sum_t FB[row,t,d] * R[row,t]
//    - softmax slice (16 x 512) staged into LDS with global_load_async_to_lds_b32
//    - both 16-wide d-tiles (full DOUT) per block -> each A fragment feeds 2 WMMAs
//    - 8 waves split K=8192; deterministic serial LDS reduction of partial C tiles
//    - diag folded into the t==15 slot (t0 is always even <= 14, so only the
//      second A element can be the diagonal slot)
// ---------------------------------------------------------------------------
__global__ void attend_k(const float* __restrict__ adj, const float* __restrict__ S,
                         const float* __restrict__ GB,  const float* __restrict__ FB,
                         const float* __restrict__ R,   float* __restrict__ OUT,
                         int transS)
{
  __shared__ float Ssh[16][NNODE];    // 32 KB softmax slice
  __shared__ float part[8][16][DDIM]; // 16 KB partial C tiles
  int row0 = blockIdx.x * 16;
  int tid = threadIdx.x, wave = tid >> 5, l = tid & 31;
  int m = l & 15, kh = l >> 4;
  int row = row0 + m;

  // stage S[row0..row0+15][*] (or its transpose) into LDS via async copy
  for (int idx = tid; idx < 16*NNODE; idx += 256) {
    int r = idx >> 9, o = idx & (NNODE - 1);
    const float* src = transS ? (S + (size_t)o*NNODE + row0 + r)
                              : (S + (size_t)(row0 + r)*NNODE + o);
    async_load_f32_to_lds((uint32_t)(uintptr_t)(&Ssh[r][o]), src);
  }
  wait_asynccnt0();
  __syncthreads();

  v8f c0 = {}, c1 = {};
  int kbeg = wave * (NTP/8), kend = kbeg + (NTP/8);
  for (int kk = kbeg; kk < kend; kk += 4) {
    int other = kk >> 4;               // same for all 4 k's of this step
    int t0    = (kk + 2*kh) & 15;      // even, <= 14
    int t1    = t0 + 1;
    if ((kk & 15) == 0)
      __builtin_prefetch(adj + ((size_t)row*NNODE + other + 16)*TT, 0, 0);
    float s = Ssh[m][other];
    const float* adjp = adj + ((size_t)row*NNODE + other)*TT;
    v2f av, b0, b1;
    av.x = s * adjp[t0];
    av.y = (t1 < TT) ? s * adjp[t1]
                     : ((row == other) ? Ssh[m][row] : 0.0f);
    const float* gb = GB + (size_t)other*TP*DDIM;
    b0.x = gb[t0*DDIM + m];        b0.y = gb[t1*DDIM + m];
    b1.x = gb[t0*DDIM + 16 + m];   b1.y = gb[t1*DDIM + 16 + m];
    c0 = wmma_f32x4(av, b0, c0);
    c1 = wmma_f32x4(av, b1, c1);
  }
#pragma unroll
  for (int r = 0; r < 8; ++r) {
    part[wave][8*kh + r][m]      = c0[r];
    part[wave][8*kh + r][16 + m] = c1[r];
  }
  __syncthreads();
  for (int idx = tid; idx < 16*DDIM; idx += 256) {
    int rr = idx >> 5, dd = idx & 31;
    float v = 0.f;
#pragma unroll
    for (int w = 0; w < 8; ++w) v += part[w][rr][dd];
    int orow = row0 + rr;
#pragma unroll
    for (int t = 0; t < TP; ++t)
      v += FB[(size_t)orow*TP*DDIM + t*DDIM + dd] * R[orow*TP + t];
    OUT[orow*DDIM + dd] = v;
  }
}

// ---------------------------------------------------------------------------
// 7) out = lrelu(temp1 + temp2)
// ---------------------------------------------------------------------------
__global__ void finalize_k(const float* __restrict__ T1, const float* __restrict__ T2,
                           float* __restrict__ out)
{
  int i = blockIdx.x * 256 + threadIdx.x;
  if (i < NNODE*DDIM) {
    float v = T1[i] + T2[i];
    out[i] = fmaxf(v, 0.2f * v);
  }
}

extern "C" void kernel_launch(void* const* d_in, const int* in_sizes, int n_in,
                              void* d_out, int out_size, void* d_ws, size_t ws_size,
                              hipStream_t stream) {
  (void)in_sizes; (void)n_in; (void)out_size; (void)ws_size;
  const float* x   = (const float*)d_in[0];
  const float* adj = (const float*)d_in[1];
  const float* W1  = (const float*)d_in[2];
  const float* W2  = (const float*)d_in[3];
  const float* W3  = (const float*)d_in[4];
  float* ws = (float*)d_ws;
  float* A1 = ws + OFF_A1; float* B1 = ws + OFF_B1;
  float* A2 = ws + OFF_A2; float* B2 = ws + OFF_B2;
  float* Q  = ws + OFF_Q;  float* M2 = ws + OFF_M2; float* M3 = ws + OFF_M3;
  float* M1 = ws + OFF_M1; float* M4 = ws + OFF_M4;
  float* L1 = ws + OFF_L1; float* L2 = ws + OFF_L2;
  float* R1 = ws + OFF_R1; float* R2 = ws + OFF_R2;
  float* T1 = ws + OFF_T1; float* T2 = ws + OFF_T2;

  precompute_k<<<NNODE, 256, 0, stream>>>(x, W1, W2, W3, A1, B1, A2, B2, Q, M2, M3);
  gemm_qat_k<<<dim3(NTP/16, NNODE/16, 2), 32, 0, stream>>>(Q, A1, B1, M1, M4);
  logits_k<<<dim3(NNODE/256, NNODE), 256, 0, stream>>>(adj, M1, M2, M3, M4, L1, L2);
  softmax_row_k<<<NNODE, 256, 0, stream>>>(L1);
  softmax_col_k<<<NNODE, 256, 0, stream>>>(L2);
  compute_r_k<<<NNODE, 256, 0, stream>>>(adj, L1, L2, R1, R2);
  attend_k<<<NNODE/16, 256, 0, stream>>>(adj, L1, A2, B2, R1, T1, 0);
  attend_k<<<NNODE/16, 256, 0, stream>>>(adj, L2, B2, A2, R2, T2, 1);
  finalize_k<<<(NNODE*DDIM + 255)/256, 256, 0, stream>>>(T1, T2, (float*)d_out);
}